// S4Layer_1477468750454
// MI455X (gfx1250) — compile-verified
//
#include <hip/hip_runtime.h>

#define DM 1024   // D_MODEL
#define DS 256    // D_STATE
#define NH 2      // NUM_HEADS
#define DH 128    // D_HEAD
#define BS 8      // BATCH
#define SL 4096   // SEQ_LEN

typedef float v2f __attribute__((ext_vector_type(2)));
typedef float v8f __attribute__((ext_vector_type(8)));

__device__ __forceinline__ v8f wmma4(v2f a, v2f b, v8f c) {
  // V_WMMA_F32_16X16X4_F32: (neg_a, A, neg_b, B, c_mod, C, reuse_a, reuse_b)
  return __builtin_amdgcn_wmma_f32_16x16x4_f32(false, a, false, b, (short)0, c,
                                               false, false);
}

// ---------------------------------------------------------------------------
// K1: bilinear discretization. M = I + (dt/2)*scale*A_hippo.
//     A_d = 2*M^-1 - I ; B_d = dt * M^-1 @ Bp.  In-place Gauss-Jordan (M~I).
//     A_d stored transposed+pair-packed as WMMA B-operand fragments:
//     adt[h][k/2][m] = float2{ A_d[m][k], A_d[m][k+1] }   (= A_dT[k..k+1][m])
// ---------------------------------------------------------------------------
__global__ void __launch_bounds__(128) k_disc(const float* __restrict__ Ah,
                                              const float* __restrict__ Bp,
                                              const float* __restrict__ logdt,
                                              const float* __restrict__ scale,
                                              float* __restrict__ adt,
                                              float* __restrict__ bd) {
  __shared__ float Ms[128 * 128];   // exactly 64 KB
  const int h = blockIdx.x;
  const int t = threadIdx.x;        // row owned by this thread
  const float dt  = expf(logdt[h]);
  const float hsc = 0.5f * dt * scale[h];
  for (int c = 0; c < 128; ++c)
    Ms[t * 128 + c] = (t == c ? 1.0f : 0.0f) + hsc * Ah[t * 128 + c];
  __syncthreads();
  for (int j = 0; j < 128; ++j) {
    const float f = Ms[t * 128 + j];   // pivot-column value before row-j scale
    if (t == j) {
      const float p = 1.0f / f;
      for (int k = 0; k < 128; ++k) Ms[j * 128 + k] *= p;
      Ms[j * 128 + j] = p;
    }
    __syncthreads();
    if (t != j) {
      const float pj = Ms[j * 128 + j];
      for (int k = 0; k < 128; ++k) Ms[t * 128 + k] -= f * Ms[j * 128 + k];
      Ms[t * 128 + j] = -f * pj;
    }
    __syncthreads();
  }
  for (int kk = 0; kk < 64; ++kk) {
    const float a0 = 2.0f * Ms[t * 128 + 2 * kk]     - (t == 2 * kk     ? 1.0f : 0.0f);
    const float a1 = 2.0f * Ms[t * 128 + 2 * kk + 1] - (t == 2 * kk + 1 ? 1.0f : 0.0f);
    *(float2*)&adt[(((size_t)h * 64 + kk) * 128 + t) * 2] = make_float2(a0, a1);
  }
  float s = 0.f;
  for (int n = 0; n < 128; ++n) s += Ms[t * 128 + n] * Bp[h * 128 + n];
  bd[h * 128 + t] = dt * s;
}

// ---------------------------------------------------------------------------
// K2: x_proj = x @ Wi^T + bi, scattered into u[h][b][t][c]  (c = col & 127).
//     One 16-row tile per WG; wave w -> output col tiles {32w, 32w+16}.
// ---------------------------------------------------------------------------
__global__ void __launch_bounds__(256) k_inproj(const float* __restrict__ x,
                                                const float* __restrict__ Wi,
                                                const float* __restrict__ bi,
                                                float* __restrict__ u) {
  const int tid = threadIdx.x, lane = tid & 31, w = tid >> 5;
  const int row0 = blockIdx.x * 16;
  const int r = lane & 15, hi = lane >> 4, koff = hi * 2;
  const int n0a = w * 32, n0b = w * 32 + 16;
  const float* xrow = x  + (size_t)(row0 + r) * DM + koff;
  const float* wra  = Wi + (size_t)(n0a  + r) * DM + koff;
  const float* wrb  = Wi + (size_t)(n0b  + r) * DM + koff;
  v8f acc0, acc1;
  const float ba = bi[n0a + r], bb = bi[n0b + r];
#pragma unroll
  for (int i = 0; i < 8; ++i) { acc0[i] = ba; acc1[i] = bb; }
#pragma unroll 4
  for (int k = 0; k < DM; k += 4) {
    const v2f af = *(const v2f*)(xrow + k);
    acc0 = wmma4(af, *(const v2f*)(wra + k), acc0);
    acc1 = wmma4(af, *(const v2f*)(wrb + k), acc1);
  }
#pragma unroll
  for (int vr = 0; vr < 8; ++vr) {
    const int grow = row0 + vr + (hi << 3);
    const int b = grow >> 12, tt = grow & (SL - 1);
    const int colA = n0a + r, colB = n0b + r;
    u[(((size_t)(colA >> 7) * BS + b) * SL + tt) * DH + (colA & 127)] = acc0[vr];
    u[(((size_t)(colB >> 7) * BS + b) * SL + tt) * DH + (colB & 127)] = acc1[vr];
  }
}

// ---------------------------------------------------------------------------
// K3: the recurrence. WG = (h, b, 16-row c-block), 8 waves.
//     State S (16x128 f32) ping-pongs in LDS (stride 132 to spread banks).
//     Wave w owns output cols m = 16w..16w+15:
//       step: D = S_frag x A_dT_frag (32 chained WMMA, K=128) + u_t (x) B_d
//       clamp to [-10,10]; y[c] = sum_m S[c][m]*Cvec[m] via shfl_xor butterfly.
//     A_dT fragments are t-invariant -> cached in 64 VGPRs before the loop.
// ---------------------------------------------------------------------------
__global__ void __launch_bounds__(256) k_scan(const float* __restrict__ u,
                                              const float* __restrict__ adt,
                                              const float* __restrict__ bd,
                                              const float* __restrict__ Cp,
                                              float* __restrict__ ys) {
  __shared__ float S[2][16 * 132];
  __shared__ float Yp[8][16];
  const int tid = threadIdx.x, lane = tid & 31, w = tid >> 5;
  const int h = blockIdx.x >> 6, b = (blockIdx.x >> 3) & 7, cb = blockIdx.x & 7;
  const int c0 = cb * 16;
  const int r = lane & 15, hi = lane >> 4, koff = hi * 2;
  const int m = w * 16 + r;
  const float bdv = bd[h * 128 + m];
  const float cvv = Cp[h * 128 + m];
  const float* gB = adt + (size_t)h * 64 * 128 * 2;
  const float* up = u + ((size_t)(h * BS + b) * SL) * DH + c0;
  float* yp = ys + (size_t)b * SL * DS + h * 128 + c0;

  // Cache all 32 t-invariant B-operand fragments of A_dT in registers.
  v2f bfr[32];
#pragma unroll
  for (int kq = 0; kq < 32; ++kq)
    bfr[kq] = *(const v2f*)(gB + ((size_t)(kq << 1) + hi) * 256 + m * 2);

  for (int i = tid; i < 2 * 16 * 132; i += 256) (&S[0][0])[i] = 0.f;
  __syncthreads();

  int cur = 0;
#pragma unroll 1
  for (int t = 0; t < SL; ++t) {
    const float uu = up[(size_t)t * DH + r];       // u_t[c0 + r], dup in halves
    v8f acc;
#pragma unroll
    for (int vr = 0; vr < 8; ++vr)                 // C = u_t[c] * B_d[m]
      acc[vr] = __shfl(uu, vr + (hi << 3), 32) * bdv;

    const float* Sp = &S[cur][0];
#pragma unroll
    for (int kq = 0; kq < 32; ++kq) {
      const v2f af = *(const v2f*)(Sp + r * 132 + (kq << 2) + koff);
      acc = wmma4(af, bfr[kq], acc);
    }

    float* Sn = &S[cur ^ 1][0];
    float pys[8];
#pragma unroll
    for (int vr = 0; vr < 8; ++vr) {
      const float v = fminf(fmaxf(acc[vr], -10.f), 10.f);
      Sn[(vr + (hi << 3)) * 132 + m] = v;          // clipped state -> next buf
      pys[vr] = v * cvv;
    }
#pragma unroll
    for (int vr = 0; vr < 8; ++vr) {               // reduce over 16-lane halves
      float p = pys[vr];
      p += __shfl_xor(p, 1, 32);
      p += __shfl_xor(p, 2, 32);
      p += __shfl_xor(p, 4, 32);
      p += __shfl_xor(p, 8, 32);
      if (r == 0) Yp[w][vr + (hi << 3)] = p;
    }
    __syncthreads();
    if (tid < 16) {
      float s = 0.f;
#pragma unroll
      for (int ww = 0; ww < 8; ++ww) s += Yp[ww][tid];
      yp[(size_t)t * DS + tid] = s;                // ys[b][t][h*128+c0+tid]
    }
    __syncthreads();
    cur ^= 1;
  }
}

// ---------------------------------------------------------------------------
// K4: out = ys @ Wo^T + bo + x  (residual), written to d_out (pre-LN).
// ---------------------------------------------------------------------------
__global__ void __launch_bounds__(256) k_outproj(const float* __restrict__ ysb,
                                                 const float* __restrict__ Wo,
                                                 const float* __restrict__ bo,
                                                 const float* __restrict__ x,
                                                 float* __restrict__ res) {
  const int tid = threadIdx.x, lane = tid & 31, w = tid >> 5;
  const int rowT = blockIdx.x >> 2, cg = blockIdx.x & 3;
  const int row0 = rowT * 16;
  const int r = lane & 15, hi = lane >> 4, koff = hi * 2;
  const int n0a = cg * 256 + w * 32, n0b = n0a + 16;
  const float* arow = ysb + (size_t)(row0 + r) * DS + koff;
  const float* wra  = Wo  + (size_t)(n0a  + r) * DS + koff;
  const float* wrb  = Wo  + (size_t)(n0b  + r) * DS + koff;
  v8f acc0, acc1;
  const float ba = bo[n0a + r], bb = bo[n0b + r];
#pragma unroll
  for (int i = 0; i < 8; ++i) { acc0[i] = ba; acc1[i] = bb; }
#pragma unroll 4
  for (int k = 0; k < DS; k += 4) {
    const v2f af = *(const v2f*)(arow + k);
    acc0 = wmma4(af, *(const v2f*)(wra + k), acc0);
    acc1 = wmma4(af, *(const v2f*)(wrb + k), acc1);
  }
#pragma unroll
  for (int vr = 0; vr < 8; ++vr) {
    const int grow = row0 + vr + (hi << 3);
    const size_t ia = (size_t)grow * DM + n0a + r;
    const size_t ib = (size_t)grow * DM + n0b + r;
    res[ia] = acc0[vr] + x[ia];
    res[ib] = acc1[vr] + x[ib];
  }
}

// ---------------------------------------------------------------------------
// K5: LayerNorm over 1024, in-place on d_out. One WG per row.
// ---------------------------------------------------------------------------
__global__ void __launch_bounds__(256) k_ln(const float* __restrict__ res,
                                            const float* __restrict__ gamma,
                                            const float* __restrict__ beta,
                                            float* __restrict__ out) {
  __shared__ float red[16];
  __shared__ float mu_s, rs_s;
  const int row = blockIdx.x, tid = threadIdx.x;
  const float* rp = res + (size_t)row * DM;
  float v[4];
  float s = 0.f, s2 = 0.f;
#pragma unroll
  for (int i = 0; i < 4; ++i) {
    v[i] = rp[tid + i * 256];
    s += v[i]; s2 += v[i] * v[i];
  }
#pragma unroll
  for (int o = 16; o >= 1; o >>= 1) {
    s  += __shfl_xor(s,  o, 32);
    s2 += __shfl_xor(s2, o, 32);
  }
  if ((tid & 31) == 0) { red[tid >> 5] = s; red[8 + (tid >> 5)] = s2; }
  __syncthreads();
  if (tid == 0) {
    float ts = 0.f, ts2 = 0.f;
    for (int i = 0; i < 8; ++i) { ts += red[i]; ts2 += red[8 + i]; }
    const float mu = ts / (float)DM;
    mu_s = mu;
    rs_s = rsqrtf(ts2 / (float)DM - mu * mu + 1e-5f);
  }
  __syncthreads();
  const float mu = mu_s, rs = rs_s;
#pragma unroll
  for (int i = 0; i < 4; ++i) {
    const int c = tid + i * 256;
    out[(size_t)row * DM + c] = (v[i] - mu) * rs * gamma[c] + beta[c];
  }
}

// ---------------------------------------------------------------------------
extern "C" void kernel_launch(void* const* d_in, const int* in_sizes, int n_in,
                              void* d_out, int out_size, void* d_ws, size_t ws_size,
                              hipStream_t stream) {
  const float* x   = (const float*)d_in[0];
  const float* Ah  = (const float*)d_in[1];
  const float* Bp  = (const float*)d_in[2];
  const float* Cp  = (const float*)d_in[3];
  const float* ldt = (const float*)d_in[4];
  const float* sc  = (const float*)d_in[5];
  const float* Wi  = (const float*)d_in[6];
  const float* bi  = (const float*)d_in[7];
  const float* Wo  = (const float*)d_in[8];
  const float* bo  = (const float*)d_in[9];
  const float* gm  = (const float*)d_in[10];
  const float* bt  = (const float*)d_in[11];
  float* out = (float*)d_out;

  char* ws = (char*)d_ws;
  float* u   = (float*)ws;                                    // H*B*L*DH  (32 MB)
  float* adt = (float*)(ws + (size_t)NH * BS * SL * DH * 4);  // H*128*128 (128 KB)
  float* bd  = (float*)((char*)adt + (size_t)NH * DH * DH * 4);
  float* ysb = (float*)((char*)bd + (size_t)NH * DH * 4);     // B*L*DS    (32 MB)

  k_disc   <<<NH,                 128, 0, stream>>>(Ah, Bp, ldt, sc, adt, bd);
  k_inproj <<<(BS * SL) / 16,     256, 0, stream>>>(x, Wi, bi, u);
  k_scan   <<<NH * BS * 8,        256, 0, stream>>>(u, adt, bd, Cp, ysb);
  k_outproj<<<((BS * SL) / 16)*4, 256, 0, stream>>>(ysb, Wo, bo, x, out);
  k_ln     <<<BS * SL,            256, 0, stream>>>(out, gm, bt, out);
}